// AdaptiveAttentionLoss_33904471835074
// MI455X (gfx1250) — compile-verified
//
#include <hip/hip_runtime.h>
#include <hip/hip_bf16.h>

#define GROUPS 4096

typedef float v2f __attribute__((ext_vector_type(2)));
typedef float v4f __attribute__((ext_vector_type(4)));
typedef float v8f __attribute__((ext_vector_type(8)));
typedef int   v4i __attribute__((ext_vector_type(4)));

// ---------------------------------------------------------------------------
// Zero the global accumulation bins (sums[G] then counts[G]) each launch.
// ---------------------------------------------------------------------------
__global__ void zero_bins_kernel(float* __restrict__ bins) {
    int i = blockIdx.x * blockDim.x + threadIdx.x;
    if (i < 2 * GROUPS) bins[i] = 0.0f;
}

// ---------------------------------------------------------------------------
// Per-sample focal-weighted CE (C=2), numerically stable:
//   d = x_other - x_true, ce = log(1+exp(d)), p = exp(-ce), w = 1 - p^2.
// ---------------------------------------------------------------------------
__device__ __forceinline__ float per_sample_loss(float x0, float x1, int lb) {
    float xt = lb ? x1 : x0;
    float xo = lb ? x0 : x1;
    float d  = xo - xt;
    float ce = fmaxf(d, 0.0f) + log1pf(__expf(-fabsf(d)));
    float p  = __expf(-ce);
    return (1.0f - p * p) * ce;
}

// ---------------------------------------------------------------------------
// Streaming pass: 4 samples / thread / iteration via B128 non-temporal loads
// (2x b128 for x, 1x b128 each for index and label = 64 B per 4 VMEM issues).
// LDS-privatized segmented sum into 4096 bins (ds_add_f32), one global
// atomic flush per block.
// ---------------------------------------------------------------------------
__global__ __launch_bounds__(1024) void focal_ce_group_kernel(
    const float* __restrict__ x,
    const int*   __restrict__ index,
    const int*   __restrict__ label,
    float*       __restrict__ bins,
    int n)
{
    __shared__ float lsum[GROUPS];
    __shared__ float lcnt[GROUPS];
    for (int g = threadIdx.x; g < GROUPS; g += blockDim.x) {
        lsum[g] = 0.0f;
        lcnt[g] = 0.0f;
    }
    __syncthreads();

    const int gid    = blockIdx.x * blockDim.x + threadIdx.x;
    const int stride = gridDim.x * blockDim.x;
    const int nvec   = n >> 2;  // quads of samples

    const v4f* xq = (const v4f*)x;      // 2 x v4f per quad (8 floats)
    const v4i* iq = (const v4i*)index;  // 1 x v4i per quad
    const v4i* lq = (const v4i*)label;

    for (int v = gid; v < nvec; v += stride) {
        v4f xa = __builtin_nontemporal_load(&xq[2 * v]);      // samples 0,1
        v4f xb = __builtin_nontemporal_load(&xq[2 * v + 1]);  // samples 2,3
        v4i gi = __builtin_nontemporal_load(&iq[v]);
        v4i li = __builtin_nontemporal_load(&lq[v]);

        float p0 = per_sample_loss(xa[0], xa[1], li[0]);
        float p1 = per_sample_loss(xa[2], xa[3], li[1]);
        float p2 = per_sample_loss(xb[0], xb[1], li[2]);
        float p3 = per_sample_loss(xb[2], xb[3], li[3]);

        atomicAdd(&lsum[gi[0]], p0);
        atomicAdd(&lcnt[gi[0]], 1.0f);
        atomicAdd(&lsum[gi[1]], p1);
        atomicAdd(&lcnt[gi[1]], 1.0f);
        atomicAdd(&lsum[gi[2]], p2);
        atomicAdd(&lcnt[gi[2]], 1.0f);
        atomicAdd(&lsum[gi[3]], p3);
        atomicAdd(&lcnt[gi[3]], 1.0f);
    }

    // Scalar tail for n % 4 != 0 (not taken for N = 16,777,216).
    for (int i = (nvec << 2) + gid; i < n; i += stride) {
        float x0 = x[2 * i], x1 = x[2 * i + 1];
        float p  = per_sample_loss(x0, x1, label[i]);
        int   g  = index[i];
        atomicAdd(&lsum[g], p);
        atomicAdd(&lcnt[g], 1.0f);
    }
    __syncthreads();

    for (int g = threadIdx.x; g < GROUPS; g += blockDim.x) {
        float cn = lcnt[g];
        if (cn != 0.0f) {
            atomicAdd(&bins[g], lsum[g]);            // global_atomic_add_f32
            atomicAdd(&bins[GROUPS + g], cn);
        }
    }
}

// ---------------------------------------------------------------------------
// Finalize: group means, present mask, average over present groups.
// Cross-lane (32-lane) reduction via one V_WMMA_F32_16X16X4_F32 per wave:
// A[16x4] carries (lsum, lpres) per lane; B[4x16] is a two-column selector
// so D column 0 = reduced sums, column 1 = reduced present-counts.
// ---------------------------------------------------------------------------
__global__ __launch_bounds__(256) void finalize_kernel(
    const float* __restrict__ bins,
    float* __restrict__ out)
{
    __shared__ float wres_sum[8];
    __shared__ float wres_pres[8];

    const int tid = threadIdx.x;
    float lsum = 0.0f, lpres = 0.0f;
    for (int g = tid; g < GROUPS; g += 256) {
        float cn = bins[GROUPS + g];
        if (cn > 0.0f) {
            lsum  += bins[g] / cn;   // group mean
            lpres += 1.0f;           // present group
        }
    }

    const int lane = tid & 31;
    // A (16x4 f32): lanes 0-15 hold K=0,1 in VGPR0,1; lanes 16-31 hold K=2,3.
    v2f a;
    a[0] = lsum;
    a[1] = lpres;
    // B (4x16 f32): VGPR0 = rows 0 (lanes 0-15) / 2 (lanes 16-31),
    //               VGPR1 = rows 1 / 3. Column 0 selects sums, column 1 pres.
    v2f b;
    b[0] = (lane == 0 || lane == 16) ? 1.0f : 0.0f;
    b[1] = (lane == 1 || lane == 17) ? 1.0f : 0.0f;
    v8f c = {};
    c = __builtin_amdgcn_wmma_f32_16x16x4_f32(
        /*neg_a=*/false, a, /*neg_b=*/false, b,
        /*c_mod=*/(short)0, c, /*reuse_a=*/false, /*reuse_b=*/false);

    // Sum this lane's 8 rows, then fold the two 8-row halves (lanes L, L+16).
    float t = c[0] + c[1] + c[2] + c[3] + c[4] + c[5] + c[6] + c[7];
    t += __shfl_xor(t, 16, 32);

    const int wid = tid >> 5;
    if (lane == 0) wres_sum[wid]  = t;  // column 0: wave total of sums
    if (lane == 1) wres_pres[wid] = t;  // column 1: wave total of present
    __syncthreads();

    if (tid == 0) {
        float s = 0.0f, pcount = 0.0f;
        for (int w = 0; w < 8; ++w) {
            s      += wres_sum[w];
            pcount += wres_pres[w];
        }
        out[0] = s / fmaxf(pcount, 1.0f);
    }
}

// ---------------------------------------------------------------------------
// Launch wrapper.
// ---------------------------------------------------------------------------
extern "C" void kernel_launch(void* const* d_in, const int* in_sizes, int n_in,
                              void* d_out, int out_size, void* d_ws, size_t ws_size,
                              hipStream_t stream) {
    const float* x     = (const float*)d_in[0];
    const int*   index = (const int*)d_in[1];
    const int*   label = (const int*)d_in[2];
    float* bins = (float*)d_ws;     // 2 * GROUPS floats = 32 KB scratch
    float* out  = (float*)d_out;
    const int n = in_sizes[0] / 2;  // x is [N, 2]

    zero_bins_kernel<<<(2 * GROUPS + 255) / 256, 256, 0, stream>>>(bins);
    focal_ce_group_kernel<<<256, 1024, 0, stream>>>(x, index, label, bins, n);
    finalize_kernel<<<1, 256, 0, stream>>>(bins, out);
}